// CostVolume_27951647162685
// MI455X (gfx1250) — compile-verified
//
#include <hip/hip_runtime.h>
#include <cstdint>

// Cost volume: out (2, 64, 48, 128, 256) fp32
//   out[b, c     , d, h, x] = mask(d,x) ? left [b, c, h, x      ] : 0
//   out[b, c + 32, d, h, x] = mask(d,x) ? right[b, c, h, x - dval] : 0
//   dval = d - 8, mask = 0 <= x - dval < 256
// Pure bandwidth problem: 805 MB of write-once output -> NT B128 stores;
// inputs staged to LDS via CDNA5 async global->LDS (ASYNCcnt) per block.

#define MIN_DISP_C (-8)
constexpr int BB = 2, CC = 32, HH = 128, WW = 256, DD = 48;
constexpr int DG  = 8;        // disparities per block
constexpr int NDG = DD / DG;  // 6

typedef float v4f __attribute__((ext_vector_type(4)));

__device__ __forceinline__ void async_load_b128(const float* gaddr, unsigned lds_byte_off) {
  // CDNA5: GLOBAL_LOAD_ASYNC_TO_LDS_B128 (GV mode). VDST = LDS byte address,
  // VADDR = 64-bit global address. Tracked with ASYNCcnt.
  asm volatile("global_load_async_to_lds_b128 %0, %1, off"
               :
               : "v"(lds_byte_off), "v"((uint64_t)(uintptr_t)gaddr)
               : "memory");
}

__global__ __launch_bounds__(256)
void cost_volume_kernel(const float* __restrict__ L,
                        const float* __restrict__ R,
                        float* __restrict__ out) {
  // 64KB staging (left rows then right rows, all 32 channels of one (b,h) line)
  // + small pad so speculative out-of-range LDS reads (discarded by select)
  // stay inside the allocation.
  __shared__ float lds[2 * CC * WW + 64];
  float* ldsL = lds;
  float* ldsR = lds + CC * WW;

  const int t  = threadIdx.x;
  const int b  = blockIdx.x / HH;
  const int h  = blockIdx.x % HH;
  const int d0 = blockIdx.y * DG;

  // Generic LDS pointer low 32 bits == byte offset within the wave's LDS.
  const unsigned lds_base = (unsigned)(uintptr_t)lds;

  const int x4 = (t & 63) * 4;  // this thread's float4 x-offset within a row
  const int ci = t >> 6;        // 0..3 channel sub-index

  // ---- stage left & right feature rows (32 channels x 256 px each) --------
  #pragma unroll
  for (int k = 0; k < 8; ++k) {
    const int c        = ci + k * 4;
    const int goff     = ((b * CC + c) * HH + h) * WW + x4;
    const unsigned off = (unsigned)((c * WW + x4) * 4);
    async_load_b128(L + goff, lds_base + off);
    async_load_b128(R + goff, lds_base + (unsigned)(CC * WW * 4) + off);
  }
  asm volatile("s_wait_asynccnt 0" ::: "memory");
  __syncthreads();

  // ---- emit 8 disparities x 64 channels for this (b, h) line --------------
  #pragma unroll 1
  for (int k = 0; k < 8; ++k) {
    const int c  = ci + k * 4;
    const v4f lv = *(const v4f*)&ldsL[c * WW + x4];
    #pragma unroll
    for (int dd = 0; dd < DG; ++dd) {
      const int d    = d0 + dd;
      const int dval = d + MIN_DISP_C;
      v4f lo, ro;
      #pragma unroll
      for (int j = 0; j < 4; ++j) {
        const int  src = x4 + j - dval;
        const bool ok  = (unsigned)src < (unsigned)WW;
        lo[j] = ok ? lv[j] : 0.0f;
        ro[j] = ok ? ldsR[c * WW + src] : 0.0f;  // misaligned gather from LDS
      }
      // out[b][ch][d][h][x], ch-left = c, ch-right = c + 32
      const size_t obL = ((((size_t)b * 64 + (size_t)c       ) * DD + d) * HH + h) * WW + x4;
      const size_t obR = ((((size_t)b * 64 + (size_t)(c + 32)) * DD + d) * HH + h) * WW + x4;
      // Write-once 805MB stream: non-temporal so it doesn't flush the
      // L2-resident (16MB, 48x-reused) inputs.
      __builtin_nontemporal_store(lo, (v4f*)(out + obL));
      __builtin_nontemporal_store(ro, (v4f*)(out + obR));
    }
  }
}

extern "C" void kernel_launch(void* const* d_in, const int* in_sizes, int n_in,
                              void* d_out, int out_size, void* d_ws, size_t ws_size,
                              hipStream_t stream) {
  const float* L = (const float*)d_in[0];   // left_feat  (2,32,128,256) fp32
  const float* R = (const float*)d_in[1];   // right_feat (2,32,128,256) fp32
  float* out = (float*)d_out;               // (2,64,48,128,256) fp32

  dim3 grid(BB * HH, NDG);                  // (256, 6) blocks
  cost_volume_kernel<<<grid, 256, 0, stream>>>(L, R, out);
}